// GINClassifier_53996328846048
// MI455X (gfx1250) — compile-verified
//
#include <hip/hip_runtime.h>

// ---------------------------------------------------------------------------
// GIN classifier on MI455X (gfx1250, wave32).
//  - 9 node GEMMs [N x K] @ [K x 128] via V_WMMA_F32_16X16X4_F32 (exact f32).
//  - Edge aggregation / mean-pool via global_atomic_add_f32 (streaming bound).
//  - One wave owns a 16x128 output strip: 8 v8f accumulators, A fragment
//    reused across the 8 N-tiles; weights stay hot in L2/WGP$.
// ---------------------------------------------------------------------------

typedef __attribute__((ext_vector_type(2))) float v2f;
typedef __attribute__((ext_vector_type(8))) float v8f;

#define N_NODES  150000
#define N_EDGES  600000
#define F_INPUT  32
#define HID      128
#define N_GRAPHS 2048

// ---- simple float4 copy --------------------------------------------------
__global__ void k_copy4(const float4* __restrict__ src, float4* __restrict__ dst,
                        long n4) {
    long i = (long)blockIdx.x * blockDim.x + threadIdx.x;
    if (i < n4) dst[i] = src[i];
}

// ---- zero fill -----------------------------------------------------------
__global__ void k_zero(float* __restrict__ p, long n) {
    long i = (long)blockIdx.x * blockDim.x + threadIdx.x;
    if (i < n) p[i] = 0.0f;
}

// ---- edge scatter-add: agg[dst] += h[src]  (one thread = edge x 4 feats) --
__global__ void k_edge_scatter(const float* __restrict__ h,
                               const int* __restrict__ ei,   // [2,E] flat
                               float* __restrict__ agg,
                               int F, int chunk_shift) {
    long gid = (long)blockIdx.x * blockDim.x + threadIdx.x;
    long total = (long)N_EDGES << chunk_shift;          // E * (F/4)
    if (gid >= total) return;
    int e = (int)(gid >> chunk_shift);
    int c = (int)(gid & ((1 << chunk_shift) - 1)) * 4;
    int src = ei[e];
    int dst = ei[N_EDGES + e];
    const float4 v = *(const float4*)(h + (long)src * F + c);
    float* p = agg + (long)dst * F + c;
    atomicAdd(p + 0, v.x);
    atomicAdd(p + 1, v.y);
    atomicAdd(p + 2, v.z);
    atomicAdd(p + 3, v.w);
}

// ---- WMMA GEMM: C[M x 128] = relu(A[M x K] @ W[K x 128] + bias) ----------
// One wave computes a 16-row x 128-col strip. M must be a multiple of 16
// (150000 = 9375*16, 2048 = 128*16).
__global__ void __launch_bounds__(128)
k_wmma_gemm_bias_relu(const float* __restrict__ A,
                      const float* __restrict__ W,     // [K x 128] row-major
                      const float* __restrict__ bias,  // [128]
                      float* __restrict__ C,
                      int M, int K) {
    const int lane = threadIdx.x & 31;
    const int wave = threadIdx.x >> 5;
    const int rt = blockIdx.x * (blockDim.x >> 5) + wave;   // row-tile
    if (rt * 16 >= M) return;     // wave-uniform: EXEC stays all-ones

    const int half = lane >> 4;   // 0: K pair {k,k+1}, 1: K pair {k+2,k+3}
    const int mn   = lane & 15;   // M index for A frag, N index for B frag

    const v8f vzero = {0.f, 0.f, 0.f, 0.f, 0.f, 0.f, 0.f, 0.f};
    v8f acc[8];
#pragma unroll
    for (int t = 0; t < 8; ++t) acc[t] = vzero;

    // A fragment base: row (rt*16 + mn), K offset 2*half
    const float* arow = A + (long)(rt * 16 + mn) * K + 2 * half;

    for (int k = 0; k < K; k += 4) {
        v2f a;
        a.x = arow[k];          // A[m][k + 2*half]
        a.y = arow[k + 1];      // A[m][k + 2*half + 1]
        const float* wr0 = W + (long)(k + 2 * half) * HID;  // row k+2*half
        const float* wr1 = wr0 + HID;                       // row k+2*half+1
#pragma unroll
        for (int t = 0; t < 8; ++t) {
            v2f b;
            b.x = wr0[t * 16 + mn];
            b.y = wr1[t * 16 + mn];
            // D = A(16x4) * B(4x16) + C ; 8 args per probe-confirmed pattern
            acc[t] = __builtin_amdgcn_wmma_f32_16x16x4_f32(
                false, a, false, b, (short)0, acc[t], false, false);
        }
    }

    // Epilogue: C VGPR j holds row (j + 8*half), col mn of each 16x16 tile.
#pragma unroll
    for (int t = 0; t < 8; ++t) {
        const float bv = bias[t * 16 + mn];
#pragma unroll
        for (int j = 0; j < 8; ++j) {
            const int row = rt * 16 + j + 8 * half;
            float v = acc[t][j] + bv;
            v = v > 0.0f ? v : 0.0f;
            C[(long)row * HID + t * 16 + mn] = v;
        }
    }
}

// ---- mean-pool accumulate: sums[batch[n]] += h[n]; cnts[batch[n]] += 1 ---
__global__ void k_pool_acc(const float* __restrict__ h,
                           const int* __restrict__ batch,
                           float* __restrict__ sums,
                           float* __restrict__ cnts) {
    long gid = (long)blockIdx.x * blockDim.x + threadIdx.x;
    const int chunks = HID / 4;                // 32
    long total = (long)N_NODES * chunks;
    if (gid >= total) return;
    int node = (int)(gid >> 5);
    int c = (int)(gid & 31) * 4;
    int b = batch[node];
    const float4 v = *(const float4*)(h + (long)node * HID + c);
    float* p = sums + (long)b * HID + c;
    atomicAdd(p + 0, v.x);
    atomicAdd(p + 1, v.y);
    atomicAdd(p + 2, v.z);
    atomicAdd(p + 3, v.w);
    if (c == 0) atomicAdd(cnts + b, 1.0f);
}

// ---- mean: g = sums / max(cnts, 1) --------------------------------------
__global__ void k_mean(const float* __restrict__ sums,
                       const float* __restrict__ cnts,
                       float* __restrict__ g) {
    int i = blockIdx.x * blockDim.x + threadIdx.x;
    if (i >= N_GRAPHS * HID) return;
    float c = cnts[i >> 7];      // i / HID
    c = c > 1.0f ? c : 1.0f;
    g[i] = sums[i] / c;
}

// ---- final head: out[g, o] = g[g] . out_w[:, o] + out_b[o]  (o in {0,1}) -
__global__ void k_head(const float* __restrict__ g,
                       const float* __restrict__ ow,   // [128 x 2]
                       const float* __restrict__ ob,   // [2]
                       float* __restrict__ out) {
    int gid = blockIdx.x * blockDim.x + threadIdx.x;
    if (gid >= N_GRAPHS * 2) return;
    int gr = gid >> 1, o = gid & 1;
    float s = ob[o];
    const float* row = g + (long)gr * HID;
#pragma unroll 4
    for (int k = 0; k < HID; ++k) s += row[k] * ow[k * 2 + o];
    out[gid] = s;
}

// ---------------------------------------------------------------------------
extern "C" void kernel_launch(void* const* d_in, const int* in_sizes, int n_in,
                              void* d_out, int out_size, void* d_ws, size_t ws_size,
                              hipStream_t stream) {
    const float* x     = (const float*)d_in[0];
    const int*   ei    = (const int*)d_in[1];
    const int*   batch = (const int*)d_in[2];
    // GIN layer weights: d_in[3 + 6*L + {0..5}] = w1,b1,w2,b2,w3,b3
    const float* lw[3][3];
    const float* lb[3][3];
    for (int L = 0; L < 3; ++L)
        for (int j = 0; j < 3; ++j) {
            lw[L][j] = (const float*)d_in[3 + 6 * L + 2 * j];
            lb[L][j] = (const float*)d_in[3 + 6 * L + 2 * j + 1];
        }
    const float* fc0_w = (const float*)d_in[21];
    const float* fc0_b = (const float*)d_in[22];
    const float* fc1_w = (const float*)d_in[23];
    const float* fc1_b = (const float*)d_in[24];
    const float* out_w = (const float*)d_in[25];
    const float* out_b = (const float*)d_in[26];
    float* out = (float*)d_out;

    // Workspace carve-up (all f32):
    float* bufA = (float*)d_ws;                                   // N*HID
    float* bufB = bufA + (size_t)N_NODES * HID;                   // N*HID
    float* sums = bufB + (size_t)N_NODES * HID;                   // G*HID
    float* cnts = sums + (size_t)N_GRAPHS * HID;                  // G
    float* g0   = cnts + N_GRAPHS;                                // G*HID
    float* g1   = g0 + (size_t)N_GRAPHS * HID;                    // G*HID

    const int TPB = 256;
    const int gemm_tpb = 128;                       // 4 waves / block
    const int node_tiles = N_NODES / 16;            // 9375
    const int node_blocks = (node_tiles + 3) / 4;
    const int g_tiles = N_GRAPHS / 16;              // 128
    const int g_blocks = (g_tiles + 3) / 4;

    // ---------------- GIN layer 0 (input feat = 32) ----------------
    {
        long n4 = (long)N_NODES * F_INPUT / 4;
        k_copy4<<<(int)((n4 + TPB - 1) / TPB), TPB, 0, stream>>>(
            (const float4*)x, (float4*)bufA, n4);
        long et = (long)N_EDGES * (F_INPUT / 4);
        k_edge_scatter<<<(int)((et + TPB - 1) / TPB), TPB, 0, stream>>>(
            x, ei, bufA, F_INPUT, 3);
        k_wmma_gemm_bias_relu<<<node_blocks, gemm_tpb, 0, stream>>>(
            bufA, lw[0][0], lb[0][0], bufB, N_NODES, F_INPUT);
        k_wmma_gemm_bias_relu<<<node_blocks, gemm_tpb, 0, stream>>>(
            bufB, lw[0][1], lb[0][1], bufA, N_NODES, HID);
        k_wmma_gemm_bias_relu<<<node_blocks, gemm_tpb, 0, stream>>>(
            bufA, lw[0][2], lb[0][2], bufB, N_NODES, HID);
    }

    // ---------------- GIN layers 1, 2 (feat = 128) ------------------
    for (int L = 1; L < 3; ++L) {
        long n4 = (long)N_NODES * HID / 4;
        k_copy4<<<(int)((n4 + TPB - 1) / TPB), TPB, 0, stream>>>(
            (const float4*)bufB, (float4*)bufA, n4);
        long et = (long)N_EDGES * (HID / 4);
        k_edge_scatter<<<(int)((et + TPB - 1) / TPB), TPB, 0, stream>>>(
            bufB, ei, bufA, HID, 5);
        k_wmma_gemm_bias_relu<<<node_blocks, gemm_tpb, 0, stream>>>(
            bufA, lw[L][0], lb[L][0], bufB, N_NODES, HID);
        k_wmma_gemm_bias_relu<<<node_blocks, gemm_tpb, 0, stream>>>(
            bufB, lw[L][1], lb[L][1], bufA, N_NODES, HID);
        k_wmma_gemm_bias_relu<<<node_blocks, gemm_tpb, 0, stream>>>(
            bufA, lw[L][2], lb[L][2], bufB, N_NODES, HID);
    }

    // ---------------- global mean pool + classifier head ------------
    {
        long nz = (long)N_GRAPHS * HID + N_GRAPHS;   // sums + cnts contiguous
        k_zero<<<(int)((nz + TPB - 1) / TPB), TPB, 0, stream>>>(sums, nz);
        long pt = (long)N_NODES * (HID / 4);
        k_pool_acc<<<(int)((pt + TPB - 1) / TPB), TPB, 0, stream>>>(
            bufB, batch, sums, cnts);
        int mt = N_GRAPHS * HID;
        k_mean<<<(mt + TPB - 1) / TPB, TPB, 0, stream>>>(sums, cnts, g0);
        k_wmma_gemm_bias_relu<<<g_blocks, gemm_tpb, 0, stream>>>(
            g0, fc0_w, fc0_b, g1, N_GRAPHS, HID);
        k_wmma_gemm_bias_relu<<<g_blocks, gemm_tpb, 0, stream>>>(
            g1, fc1_w, fc1_b, g0, N_GRAPHS, HID);
        int ht = N_GRAPHS * 2;
        k_head<<<(ht + TPB - 1) / TPB, TPB, 0, stream>>>(g0, out_w, out_b, out);
    }
}